// CNN_53472342835791
// MI455X (gfx1250) — compile-verified
//
#include <hip/hip_runtime.h>
#include <hip/hip_bf16.h>

// ---------------------------------------------------------------------------
// SE(3) CNN forward for MI455X (gfx1250).
// Convs are implicit GEMMs on the WMMA pipe using V_WMMA_F32_16X16X4_F32
// (full fp32 precision, matching the fp32 reference).
//
// K is ordered (ci, kz, ky, kx padded 7->8) so the loop nest is division-
// free.  Activation buffers carry 16384-float guard bands on both sides, so
// boundary reads need no clamping: loads are unconditional (pair loads at
// immediate offsets off a per-lane row pointer) and invalid elements are
// zeroed by selects, keeping EXEC all-1s around every WMMA.
// Each wave computes a 64(M) x 16(N) tile (4 accumulators): the B fragment
// is loaded once per weight row and reused by all 4 subtiles.
// ---------------------------------------------------------------------------

typedef float v2f  __attribute__((ext_vector_type(2)));
typedef float v8f  __attribute__((ext_vector_type(8)));
typedef float v2fu __attribute__((ext_vector_type(2), aligned(4)));

#define EPSF 1e-5f
#define GUARDF 16384          // guard floats on each side of activation buffers

// ---------------------------------------------------------------------------
// small utility kernels
// ---------------------------------------------------------------------------

__global__ void zero_stats(float* stats, int n) {
    int i = blockIdx.x * blockDim.x + threadIdx.x;
    if (i < n) stats[i] = 0.0f;
}

// input BN: accumulate sum and sum-of-squares over the whole tensor
__global__ void reduce_sum_sq(const float* __restrict__ x, float* stats, long long n) {
    __shared__ float ss[2];
    if (threadIdx.x == 0) { ss[0] = 0.0f; ss[1] = 0.0f; }
    __syncthreads();
    float s = 0.0f, q = 0.0f;
    for (long long i = blockIdx.x * (long long)blockDim.x + threadIdx.x; i < n;
         i += (long long)gridDim.x * blockDim.x) {
        float v = x[i];
        s += v; q += v * v;
    }
    atomicAdd(&ss[0], s);
    atomicAdd(&ss[1], q);
    __syncthreads();
    if (threadIdx.x == 0) {
        atomicAdd(&stats[0], ss[0]);
        atomicAdd(&stats[1], ss[1]);
    }
}

__global__ void norm_input(const float* __restrict__ x, float* __restrict__ y,
                           const float* stats, long long n) {
    float mu  = stats[0] / (float)n;
    float var = stats[1] / (float)n - mu * mu;
    float rs  = rsqrtf(var + EPSF);
    for (long long i = blockIdx.x * (long long)blockDim.x + threadIdx.x; i < n;
         i += (long long)gridDim.x * blockDim.x) {
        y[i] = (x[i] - mu) * rs;
    }
}

// ---------------------------------------------------------------------------
// tensor-product expansion: append 9 channels (outer square) per l=1 field
// ---------------------------------------------------------------------------
__global__ void tp_expand(const float* __restrict__ x, float* __restrict__ y,
                          int B, int C, int bm, int a, long long S3) {
    int Cout = C + 9 * bm;
    long long total = (long long)B * Cout * S3;
    for (long long idx = blockIdx.x * (long long)blockDim.x + threadIdx.x; idx < total;
         idx += (long long)gridDim.x * blockDim.x) {
        long long s  = idx % S3;
        int ch       = (int)((idx / S3) % Cout);
        int bb       = (int)(idx / (S3 * Cout));
        if (ch < C) {
            y[idx] = x[((long long)bb * C + ch) * S3 + s];
        } else {
            int t = ch - C;
            int m = t / 9;
            int ij = t % 9;
            int i = ij / 3, j = ij % 3;
            float vi = x[((long long)bb * C + a + 3 * m + i) * S3 + s];
            float vj = x[((long long)bb * C + a + 3 * m + j) * S3 + s];
            y[idx] = vi * vj;
        }
    }
}

// ---------------------------------------------------------------------------
// per-irrep batch norm:  var[f] = sum(x^2 over batch,dim,spatial) / (B*S3*d)
// ---------------------------------------------------------------------------
__device__ __forceinline__ void field_of_channel(int ch, int a, int b, int& f, int& d) {
    if (ch < a)            { f = ch;                       d = 1; }
    else if (ch < a + 3*b) { f = a + (ch - a) / 3;         d = 3; }
    else                   { f = a + b + (ch - a - 3*b)/5; d = 5; }
}

__global__ void bn_reduce_sq(const float* __restrict__ x, float* stats,
                             int B, int C, long long S3, int a, int b, int c) {
    __shared__ float sacc[64];
    int F = a + b + c;
    for (int i = threadIdx.x; i < F; i += blockDim.x) sacc[i] = 0.0f;
    __syncthreads();
    long long total = (long long)B * C * S3;
    for (long long idx = blockIdx.x * (long long)blockDim.x + threadIdx.x; idx < total;
         idx += (long long)gridDim.x * blockDim.x) {
        int ch = (int)((idx / S3) % C);
        int f, d;
        field_of_channel(ch, a, b, f, d);
        float v = x[idx];
        atomicAdd(&sacc[f], v * v);
    }
    __syncthreads();
    for (int i = threadIdx.x; i < F; i += blockDim.x) atomicAdd(&stats[i], sacc[i]);
}

__global__ void bn_normalize(float* __restrict__ x, const float* stats,
                             int B, int C, long long S3, int a, int b, int c,
                             const float* __restrict__ bias, int relu) {
    long long total = (long long)B * C * S3;
    for (long long idx = blockIdx.x * (long long)blockDim.x + threadIdx.x; idx < total;
         idx += (long long)gridDim.x * blockDim.x) {
        int ch = (int)((idx / S3) % C);
        int f, d;
        field_of_channel(ch, a, b, f, d);
        float var = stats[f] / ((float)B * (float)S3 * (float)d);
        float v = x[idx] * rsqrtf(var + EPSF);
        if (relu && ch < a) {
            v = fmaxf(v + bias[ch], 0.0f);
        }
        x[idx] = v;
    }
}

// ---------------------------------------------------------------------------
// 3-D conv (7x7x7, pad 3) as implicit GEMM on WMMA f32 16x16x4.
//   M = B * oS^3 (multiple of 64 for all layers), N = Cout, K' = Cin*7*7*8
// A fragment (16x4 f32): lane&15 = M row; lanes>=16 hold K+2,K+3; VGPR j->K+j.
// B fragment (4x16 f32): lane&15 = N col; same K striping.
// C/D (16x16 f32, 8 VGPRs): VGPR r -> M = r + 8*(lane>=16), N = lane&15.
// Input pointer must sit inside a buffer with >= GUARDF floats of slack on
// both sides (guaranteed by kernel_launch's workspace layout).
// ---------------------------------------------------------------------------
__global__ void conv3d_wmma(const float* __restrict__ in, const float* __restrict__ w,
                            float* __restrict__ out,
                            int B, int Cin, int S, int Cout, int oS, int stride) {
    const int K     = Cin * 343;
    const int lane  = threadIdx.x & 31;
    const int numNt = (Cout + 15) >> 4;
    const int tileN = blockIdx.x % numNt;
    const int tileM = blockIdx.x / numNt;       // 64-row M tile

    const int oS2 = oS * oS;
    const int oS3 = oS2 * oS;
    const long long S2 = (long long)S * S;
    const long long S3 = S2 * S;

    const int mRow  = lane & 15;
    const int kHalf = (lane >> 4) * 2;          // 0 or 2

    // per-subtile constants: base pointer (with x/k offset folded in),
    // z/y window starts, and loop-invariant x/k validity masks
    int izb[4], iyb[4];
    const float* abase[4];
    bool xok[4][2][2];
#pragma unroll
    for (int s = 0; s < 4; ++s) {
        int m = tileM * 64 + s * 16 + mRow;
        int r = m;
        int bb = r / oS3;  r -= bb * oS3;
        int oz = r / oS2;  r -= oz * oS2;
        int oy = r / oS;
        int ox = r - oy * oS;
        izb[s] = oz * stride - 3;
        iyb[s] = oy * stride - 3;
        int ixb = ox * stride - 3;
        abase[s] = in + (long long)bb * Cin * S3 + (ixb + kHalf);
#pragma unroll
        for (int q = 0; q < 2; ++q)
#pragma unroll
            for (int j = 0; j < 2; ++j) {
                int kx = kHalf + 4 * q + j;
                xok[s][q][j] = (kx < 7) && ((unsigned)(ixb + kx) < (unsigned)S);
            }
    }

    // weight-column constants (w rows are 7 floats; pair loads never cross a
    // row: q0 pair = [kHalf, kHalf+1], q1 pair = [4,5] or shifted [5,6])
    const int nB   = tileN * 16 + (lane & 15);
    const bool nOK = (nB < Cout);
    const bool bsh = (kHalf == 2);
    const int  w1off = bsh ? 3 : 4;             // q1 pair base relative to q0 base
    const float* wBase = w + (long long)(nOK ? nB : 0) * K + kHalf;

    v8f acc[4];
    const v8f vz = {0.f, 0.f, 0.f, 0.f, 0.f, 0.f, 0.f, 0.f};
#pragma unroll
    for (int s = 0; s < 4; ++s) acc[s] = vz;

    for (int ci = 0; ci < Cin; ++ci) {
        const float* wCi = wBase + ci * 343;
        for (int kz = 0; kz < 7; ++kz) {
            bool zok[4];
            const float* rowA[4];
#pragma unroll
            for (int s = 0; s < 4; ++s) {
                int iz = izb[s] + kz;
                zok[s] = (unsigned)iz < (unsigned)S;
                rowA[s] = abase[s] + ((long long)ci * S + iz) * S2 + (long long)iyb[s] * S;
            }
            const float* wp0 = wCi + kz * 49;
            const float* wp1 = wp0 + w1off;
            for (int ky = 0; ky < 7; ++ky) {
                // B fragments (shared by all 4 subtiles)
                v2fu b0p = *(const v2fu*)wp0;
                v2fu b1p = *(const v2fu*)wp1;
                float b00 = nOK ? b0p.x : 0.f;
                float b01 = nOK ? b0p.y : 0.f;
                float b10 = bsh ? b1p.y : b1p.x;
                b10 = nOK ? b10 : 0.f;
                float b11 = (nOK && !bsh) ? b1p.y : 0.f;
                const v2f bf0 = {b00, b01};
                const v2f bf1 = {b10, b11};
#pragma unroll
                for (int s = 0; s < 4; ++s) {
                    const bool rok = zok[s] && ((unsigned)(iyb[s] + ky) < (unsigned)S);
                    v2fu a0p = *(const v2fu*)(rowA[s]);
                    v2fu a1p = *(const v2fu*)(rowA[s] + 4);
                    const v2f af0 = {(rok && xok[s][0][0]) ? a0p.x : 0.f,
                                     (rok && xok[s][0][1]) ? a0p.y : 0.f};
                    const v2f af1 = {(rok && xok[s][1][0]) ? a1p.x : 0.f,
                                     (rok && xok[s][1][1]) ? a1p.y : 0.f};
                    acc[s] = __builtin_amdgcn_wmma_f32_16x16x4_f32(
                        false, af0, false, bf0, (short)0, acc[s], false, false);
                    acc[s] = __builtin_amdgcn_wmma_f32_16x16x4_f32(
                        false, af1, false, bf1, (short)0, acc[s], false, false);
                    rowA[s] += S;
                }
                wp0 += 7;
                wp1 += 7;
            }
        }
    }

    // ---- store D tiles (NCDHW output)
    if (nOK) {
#pragma unroll
        for (int s = 0; s < 4; ++s) {
#pragma unroll
            for (int rI = 0; rI < 8; ++rI) {
                int m = tileM * 64 + s * 16 + rI + 8 * (lane >> 4);
                int r = m;
                int bb = r / oS3;  r -= bb * oS3;
                int z  = r / oS2;  r -= z * oS2;
                int y  = r / oS;
                int x  = r - y * oS;
                out[((long long)bb * Cout + nB) * oS3 + (long long)z * oS2 + y * oS + x] =
                    acc[s][rI];
            }
        }
    }
}

// ---------------------------------------------------------------------------
// global spatial max-pool (B=8, C=10, S3=64) + BatchNorm1d(affine) -> d_out
// ---------------------------------------------------------------------------
__global__ void pool_bn_out(const float* __restrict__ x,
                            const float* __restrict__ gamma,
                            const float* __restrict__ beta,
                            float* __restrict__ out, int B, int C, int S3) {
    __shared__ float y[128];
    __shared__ float mu[32];
    __shared__ float rs[32];
    int t = threadIdx.x;
    if (t < B * C) {
        int bb = t / C, c = t % C;
        const float* p = x + ((long long)bb * C + c) * S3;
        float m = p[0];
        for (int i = 1; i < S3; ++i) m = fmaxf(m, p[i]);
        y[t] = m;
    }
    __syncthreads();
    if (t < C) {
        float s = 0.0f, q = 0.0f;
        for (int bb = 0; bb < B; ++bb) {
            float v = y[bb * C + t];
            s += v; q += v * v;
        }
        float m = s / (float)B;
        float var = q / (float)B - m * m;
        mu[t] = m;
        rs[t] = rsqrtf(var + EPSF);
    }
    __syncthreads();
    if (t < B * C) {
        int c = t % C;
        out[t] = (y[t] - mu[c]) * rs[c] * gamma[c] + beta[c];
    }
}

// ---------------------------------------------------------------------------
// host-side orchestration
// ---------------------------------------------------------------------------
static inline int grid1d(long long n, int cap = 4096) {
    long long b = (n + 255) / 256;
    if (b > cap) b = cap;
    if (b < 1) b = 1;
    return (int)b;
}

extern "C" void kernel_launch(void* const* d_in, const int* in_sizes, int n_in,
                              void* d_out, int out_size, void* d_ws, size_t ws_size,
                              hipStream_t stream) {
    (void)in_sizes; (void)n_in; (void)out_size; (void)ws_size;

    const float* x = (const float*)d_in[0];
    const float* w[7];
    for (int i = 0; i < 7; ++i) w[i] = (const float*)d_in[1 + i];
    const float* bias[6];
    for (int i = 0; i < 6; ++i) bias[i] = (const float*)d_in[8 + i];
    const float* gamma = (const float*)d_in[14];
    const float* beta  = (const float*)d_in[15];

    // workspace layout (floats):
    // [guard][bufA data][guard][bufB data][guard][stats]
    const size_t MAXE = 19660800;            // 8 * 75 * 32^3 (largest activation)
    float* wsf   = (float*)d_ws;
    float* bufA  = wsf + GUARDF;
    float* bufB  = bufA + MAXE + GUARDF;
    float* stats = bufB + MAXE + GUARDF;     // 64 floats

    // irrep features (a, b, c) and strides from the reference
    const int FA[8] = {1, 4, 10, 10, 12, 12, 14, 10};
    const int FB[8] = {0, 2,  5,  5,  6,  6,  7,  0};
    const int FC[8] = {0, 0,  1,  1,  1,  2,  2,  0};
    const int STR[7] = {2, 1, 2, 1, 2, 1, 2};
    const int Bn = 8;

    // ---- input batch norm (single channel, full-tensor stats) into bufA
    long long Nx = (long long)Bn * 64 * 64 * 64;
    zero_stats<<<1, 64, 0, stream>>>(stats, 64);
    reduce_sum_sq<<<grid1d(Nx), 256, 0, stream>>>(x, stats, Nx);
    norm_input<<<grid1d(Nx), 256, 0, stream>>>(x, bufA, stats, Nx);

    float* cur = bufA;
    float* oth = bufB;
    int S = 64;

    for (int i = 0; i < 7; ++i) {
        const int a  = FA[i], bm = FB[i], cc = FC[i];
        const int C  = a + 3 * bm + 5 * cc;
        long long S3 = (long long)S * S * S;

        // tensor-product channel expansion
        int Cin = C;
        if (bm > 0) {
            Cin = C + 9 * bm;
            long long tot = (long long)Bn * Cin * S3;
            tp_expand<<<grid1d(tot), 256, 0, stream>>>(cur, oth, Bn, C, bm, a, S3);
            float* t = cur; cur = oth; oth = t;
        }

        // conv (7x7x7, pad 3) via WMMA implicit GEMM
        const int stride = STR[i];
        const int oS = (S - 1) / stride + 1;
        const int a2 = FA[i + 1], b2 = FB[i + 1], c2 = FC[i + 1];
        const int Cout = a2 + 3 * b2 + 5 * c2;
        const long long M = (long long)Bn * oS * oS * oS;   // multiple of 64
        const int numMt = (int)(M / 64);
        const int numNt = (Cout + 15) / 16;
        conv3d_wmma<<<dim3(numMt * numNt), dim3(32), 0, stream>>>(
            cur, w[i], oth, Bn, Cin, S, Cout, oS, stride);
        { float* t = cur; cur = oth; oth = t; }
        S = oS;
        S3 = (long long)S * S * S;

        // per-irrep batch norm (+ bias/ReLU on scalar channels for i < 6)
        long long tot = (long long)Bn * Cout * S3;
        zero_stats<<<1, 64, 0, stream>>>(stats, 64);
        bn_reduce_sq<<<grid1d(tot), 256, 0, stream>>>(cur, stats, Bn, Cout, S3, a2, b2, c2);
        bn_normalize<<<grid1d(tot), 256, 0, stream>>>(
            cur, stats, Bn, Cout, S3, a2, b2, c2,
            (i < 6) ? bias[i] : (const float*)nullptr, (i < 6) ? 1 : 0);
    }

    // ---- global max pool + BatchNorm1d(affine) -> d_out (8 x 10)
    pool_bn_out<<<1, 128, 0, stream>>>(cur, gamma, beta, (float*)d_out,
                                       Bn, 10, S * S * S);
}